// Net_41223096107028
// MI455X (gfx1250) — compile-verified
//
#include <hip/hip_runtime.h>
#include <math.h>

// ---------------------------------------------------------------------------
// Problem constants (from reference): B=4, T=512 -> NTOK=2048 tokens,
// IDIM=128, HDIM=1024, CDIM=256, n_iter=4, IGNORE=10000.0f
// ---------------------------------------------------------------------------
#define NTOK   2048
#define IDIM   128
#define CDIM   256
#define NITER  4
#define SSH    255           // 2*IDIM-1 shifts
#define PADW   382           // 3*IDIM-2 padded width
#define IGNORE_OUT 10000.0f

typedef float v2f __attribute__((ext_vector_type(2)));
typedef float v8f __attribute__((ext_vector_type(8)));

// ---------------------------------------------------------------------------
// init: zero the loss accumulator and the unmasked-element counter
// ---------------------------------------------------------------------------
__global__ void init_kernel(float* loss_total, unsigned* denom) {
    *loss_total = 0.0f;
    *denom = 0u;
}

// ---------------------------------------------------------------------------
// copy x->x_res, y->y_res; count elements with y != IGNORE_OUT (integer
// atomics: order-independent -> deterministic)
// ---------------------------------------------------------------------------
__global__ void copy_count_kernel(const float* __restrict__ x,
                                  const float* __restrict__ y,
                                  float* __restrict__ x_res,
                                  float* __restrict__ y_res,
                                  unsigned* __restrict__ denom, int n) {
    int i = blockIdx.x * blockDim.x + threadIdx.x;
    bool pred = false;
    if (i < n) {
        x_res[i] = x[i];
        float yv = y[i];
        y_res[i] = yv;
        pred = (yv != IGNORE_OUT);
    }
    unsigned long long b = __ballot(pred);
    if ((threadIdx.x & (warpSize - 1)) == 0) {
        atomicAdd(denom, (unsigned)__popcll(b));
    }
}

// ---------------------------------------------------------------------------
// argaug: per-token sliding cosine correlation + argmax (first max),
// softmax attention, reverse shift, x_res residual update.
// One block of 128 threads per token.
// ---------------------------------------------------------------------------
__global__ void argaug_kernel(float* __restrict__ x_res,
                              const float* __restrict__ y_res,
                              float* __restrict__ x_attn_out) {
    __shared__ float xp[PADW + 2];   // zero-padded x_res row
    __shared__ float yv[IDIM];
    __shared__ float sim[SSH + 1];
    __shared__ float rv[IDIM];
    __shared__ int   ri[IDIM];
    __shared__ float xa[IDIM];

    const int t   = blockIdx.x;
    const int tid = threadIdx.x;

    for (int j = tid; j < PADW + 2; j += IDIM) xp[j] = 0.0f;
    __syncthreads();
    xp[(IDIM - 1) + tid] = x_res[t * IDIM + tid];
    float yval = y_res[t * IDIM + tid];
    yv[tid] = yval;

    // ||y||^2 reduction
    rv[tid] = yval * yval;
    __syncthreads();
    for (int off = 64; off > 0; off >>= 1) {
        if (tid < off) rv[tid] += rv[tid + off];
        __syncthreads();
    }
    float ynorm2 = rv[0];
    __syncthreads();

    // sliding correlation: sim[s] = <y, xp[s..s+127]> / (||y||*||slice||), 0 if denom==0
    for (int s = tid; s < SSH; s += IDIM) {
        float num = 0.0f, den2 = 0.0f;
        for (int d = 0; d < IDIM; ++d) {
            float v = xp[s + d];
            num  = fmaf(yv[d], v, num);
            den2 = fmaf(v, v, den2);
        }
        sim[s] = (den2 == 0.0f || ynorm2 == 0.0f)
                   ? 0.0f
                   : num / sqrtf(den2 * ynorm2);
    }
    __syncthreads();

    // argmax (first occurrence of max -> smaller index wins ties)
    float bv = sim[tid];
    int   bi = tid;
    if (tid + IDIM < SSH) {
        float v2 = sim[tid + IDIM];
        if (v2 > bv) { bv = v2; bi = tid + IDIM; }
    }
    rv[tid] = bv; ri[tid] = bi;
    __syncthreads();
    for (int off = 64; off > 0; off >>= 1) {
        if (tid < off) {
            float ov = rv[tid + off]; int oi = ri[tid + off];
            if (ov > rv[tid] || (ov == rv[tid] && oi < ri[tid])) {
                rv[tid] = ov; ri[tid] = oi;
            }
        }
        __syncthreads();
    }
    int idx = ri[0];
    __syncthreads();

    // x_aug = selected slice; attn = softmax(x_aug * y); x_attn = x_aug * attn
    float xaug = xp[idx + tid];
    float tv = xaug * yval;
    rv[tid] = tv;
    __syncthreads();
    for (int off = 64; off > 0; off >>= 1) {
        if (tid < off) rv[tid] = fmaxf(rv[tid], rv[tid + off]);
        __syncthreads();
    }
    float mx = rv[0];
    __syncthreads();
    float e = expf(tv - mx);
    rv[tid] = e;
    __syncthreads();
    for (int off = 64; off > 0; off >>= 1) {
        if (tid < off) rv[tid] += rv[tid + off];
        __syncthreads();
    }
    float ssum = rv[0];
    float xattn = xaug * (e / ssum);
    xa[tid] = xattn;
    x_attn_out[t * IDIM + tid] = xattn;
    __syncthreads();

    // reverse shift: x_ele[d] = x_attn[d + 127 - idx] (in range), 0 otherwise
    int src = tid + (IDIM - 1) - idx;
    float xele = (src >= 0 && src < IDIM) ? xa[src] : 0.0f;
    x_res[t * IDIM + tid] -= xele;
}

// ---------------------------------------------------------------------------
// GEMM1 (WMMA fp32): h[t,c] = sum_k x_attn[t,k] * w1[sbase+c,k] + b1[sbase+c]
// M=NTOK, N=CDIM, K=IDIM. One wave32 per 16(M)x64(N) output macro-tile:
// 4 independent accumulator chains reuse each A fragment 4x (better XDL
// pipelining, 1.25 b64 loads per WMMA instead of 2).
// A frag (16x4 f32, 2 VGPR): lane m(0..15) row M=m; half selects K pair {2h,2h+1}
// B frag (4x16 f32, 2 VGPR): lane n(0..15) col N=n; half selects K pair {2h,2h+1}
// C/D (16x16 f32, 8 VGPR): vgpr r -> M=r+8*half, N=lane&15
// ---------------------------------------------------------------------------
__global__ void gemm1_kernel(const float* __restrict__ xa,
                             const float* __restrict__ w1,
                             const float* __restrict__ b1,
                             float* __restrict__ h, int sbase) {
    const int lane  = threadIdx.x;
    const int tileM = blockIdx.x;          // 0..127
    const int nbase = blockIdx.y * 64;     // 0,64,128,192 within slice
    const int rlo   = lane & 15;
    const int half  = lane >> 4;
    const int kk    = half * 2;

    const int gRow = tileM * 16 + rlo;     // token
    const float* ap = xa + (size_t)gRow * IDIM;
    const float* bp0 = w1 + (size_t)(sbase + nbase +  0 + rlo) * IDIM;
    const float* bp1 = w1 + (size_t)(sbase + nbase + 16 + rlo) * IDIM;
    const float* bp2 = w1 + (size_t)(sbase + nbase + 32 + rlo) * IDIM;
    const float* bp3 = w1 + (size_t)(sbase + nbase + 48 + rlo) * IDIM;

    v8f acc0 = {0.f,0.f,0.f,0.f,0.f,0.f,0.f,0.f};
    v8f acc1 = acc0, acc2 = acc0, acc3 = acc0;

    #pragma unroll 8
    for (int k = 0; k < IDIM; k += 4) {
        v2f a, b0, b1f, b2, b3;
        a.x  = ap[k + kk];  a.y  = ap[k + kk + 1];
        b0.x = bp0[k + kk]; b0.y = bp0[k + kk + 1];
        b1f.x = bp1[k + kk]; b1f.y = bp1[k + kk + 1];
        b2.x = bp2[k + kk]; b2.y = bp2[k + kk + 1];
        b3.x = bp3[k + kk]; b3.y = bp3[k + kk + 1];
        acc0 = __builtin_amdgcn_wmma_f32_16x16x4_f32(false, a, false, b0,  (short)0, acc0, false, false);
        acc1 = __builtin_amdgcn_wmma_f32_16x16x4_f32(false, a, false, b1f, (short)0, acc1, false, false);
        acc2 = __builtin_amdgcn_wmma_f32_16x16x4_f32(false, a, false, b2,  (short)0, acc2, false, false);
        acc3 = __builtin_amdgcn_wmma_f32_16x16x4_f32(false, a, false, b3,  (short)0, acc3, false, false);
    }

    #pragma unroll
    for (int j = 0; j < 4; ++j) {
        v8f acc = (j == 0) ? acc0 : (j == 1) ? acc1 : (j == 2) ? acc2 : acc3;
        float bias = b1[sbase + nbase + j * 16 + rlo];
        #pragma unroll
        for (int r = 0; r < 8; ++r) {
            int m = r + half * 8;
            h[(size_t)(tileM * 16 + m) * CDIM + nbase + j * 16 + rlo] = acc[r] + bias;
        }
    }
}

// ---------------------------------------------------------------------------
// GEMM2 (WMMA fp32) + fused loss/residual epilogue:
// y_ele[t,o] = sum_c h[t,c] * w2[o,sbase+c] + b2[o]
// One wave per 16(M)x64(N) macro-tile (4 accumulators, A reused 4x).
// lsum += (y_ele - y_res)^2 where y_orig != IGNORE; y_res -= y_ele.
// Per-tile loss partials wave-reduced (shfl_xor, fixed order) -> partial[].
// ---------------------------------------------------------------------------
__global__ void gemm2_kernel(const float* __restrict__ hbuf,
                             const float* __restrict__ w2,
                             const float* __restrict__ b2,
                             const float* __restrict__ y_orig,
                             float* __restrict__ y_res,
                             float* __restrict__ partial, int sbase) {
    const int lane  = threadIdx.x;
    const int tileM = blockIdx.x;          // 0..127
    const int nbase = blockIdx.y * 64;     // 0 or 64
    const int rlo   = lane & 15;
    const int half  = lane >> 4;
    const int kk    = half * 2;

    const int gRow = tileM * 16 + rlo;     // token
    const float* ap  = hbuf + (size_t)gRow * CDIM;
    const float* bp0 = w2 + (size_t)(nbase +  0 + rlo) * 1024 + sbase;
    const float* bp1 = w2 + (size_t)(nbase + 16 + rlo) * 1024 + sbase;
    const float* bp2 = w2 + (size_t)(nbase + 32 + rlo) * 1024 + sbase;
    const float* bp3 = w2 + (size_t)(nbase + 48 + rlo) * 1024 + sbase;

    v8f acc0 = {0.f,0.f,0.f,0.f,0.f,0.f,0.f,0.f};
    v8f acc1 = acc0, acc2 = acc0, acc3 = acc0;

    #pragma unroll 8
    for (int k = 0; k < CDIM; k += 4) {
        v2f a, b0, b1f, b2f, b3;
        a.x  = ap[k + kk];  a.y  = ap[k + kk + 1];
        b0.x = bp0[k + kk]; b0.y = bp0[k + kk + 1];
        b1f.x = bp1[k + kk]; b1f.y = bp1[k + kk + 1];
        b2f.x = bp2[k + kk]; b2f.y = bp2[k + kk + 1];
        b3.x = bp3[k + kk]; b3.y = bp3[k + kk + 1];
        acc0 = __builtin_amdgcn_wmma_f32_16x16x4_f32(false, a, false, b0,  (short)0, acc0, false, false);
        acc1 = __builtin_amdgcn_wmma_f32_16x16x4_f32(false, a, false, b1f, (short)0, acc1, false, false);
        acc2 = __builtin_amdgcn_wmma_f32_16x16x4_f32(false, a, false, b2f, (short)0, acc2, false, false);
        acc3 = __builtin_amdgcn_wmma_f32_16x16x4_f32(false, a, false, b3,  (short)0, acc3, false, false);
    }

    float lsum = 0.0f;
    #pragma unroll
    for (int j = 0; j < 4; ++j) {
        v8f acc = (j == 0) ? acc0 : (j == 1) ? acc1 : (j == 2) ? acc2 : acc3;
        float bias = b2[nbase + j * 16 + rlo];
        #pragma unroll
        for (int r = 0; r < 8; ++r) {
            int m = r + half * 8;
            int o = (tileM * 16 + m) * IDIM + nbase + j * 16 + rlo;
            float ye = acc[r] + bias;
            float yr = y_res[o];
            float d  = ye - yr;
            if (y_orig[o] != IGNORE_OUT) lsum += d * d;
            y_res[o] = yr - ye;
        }
    }
    // deterministic wave32 reduction
    for (int m = 16; m > 0; m >>= 1) lsum += __shfl_xor(lsum, m, 32);
    if (lane == 0) partial[blockIdx.y * gridDim.x + blockIdx.x] = lsum;
}

// ---------------------------------------------------------------------------
// deterministic fixed-order reduction of per-tile partials into loss_total
// ---------------------------------------------------------------------------
__global__ void reduce_partial_kernel(const float* __restrict__ partial, int n,
                                      float* __restrict__ loss_total) {
    __shared__ float buf[256];
    int tid = threadIdx.x;
    float s = 0.0f;
    for (int j = tid; j < n; j += 256) s += partial[j];
    buf[tid] = s;
    __syncthreads();
    for (int off = 128; off > 0; off >>= 1) {
        if (tid < off) buf[tid] += buf[tid + off];
        __syncthreads();
    }
    if (tid == 0) *loss_total += buf[0];
}

__global__ void finalize_kernel(const float* __restrict__ loss_total,
                                const unsigned* __restrict__ denom,
                                float* __restrict__ out) {
    out[0] = loss_total[0] / ((float)NITER * (float)denom[0]);
}

// ---------------------------------------------------------------------------
// host launcher
// ---------------------------------------------------------------------------
extern "C" void kernel_launch(void* const* d_in, const int* in_sizes, int n_in,
                              void* d_out, int out_size, void* d_ws, size_t ws_size,
                              hipStream_t stream) {
    const float* x  = (const float*)d_in[0];
    const float* y  = (const float*)d_in[1];
    const float* w1 = (const float*)d_in[2];
    const float* b1 = (const float*)d_in[3];
    const float* w2 = (const float*)d_in[4];
    const float* b2 = (const float*)d_in[5];
    float* out = (float*)d_out;

    // workspace layout (floats)
    float* ws      = (float*)d_ws;
    float* x_res   = ws;                                   // 2048*128
    float* y_res   = ws + (size_t)NTOK * IDIM;             // 2048*128
    float* x_attn  = ws + (size_t)2 * NTOK * IDIM;         // 2048*128
    float* hbuf    = ws + (size_t)3 * NTOK * IDIM;         // 2048*256
    float* partial = hbuf + (size_t)NTOK * CDIM;           // 256
    float* loss_t  = partial + 256;                        // 1
    unsigned* denom = (unsigned*)(loss_t + 1);             // 1

    const int NELT = NTOK * IDIM;

    init_kernel<<<1, 1, 0, stream>>>(loss_t, denom);
    copy_count_kernel<<<(NELT + 255) / 256, 256, 0, stream>>>(x, y, x_res, y_res,
                                                              denom, NELT);
    for (int i = 0; i < NITER; ++i) {
        int sbase = i * CDIM;
        argaug_kernel<<<NTOK, IDIM, 0, stream>>>(x_res, y_res, x_attn);
        gemm1_kernel<<<dim3(NTOK / 16, CDIM / 64), 32, 0, stream>>>(
            x_attn, w1, b1, hbuf, sbase);
        gemm2_kernel<<<dim3(NTOK / 16, IDIM / 64), 32, 0, stream>>>(
            hbuf, w2, b2, y, y_res, partial, sbase);
        reduce_partial_kernel<<<1, 256, 0, stream>>>(
            partial, (NTOK / 16) * (IDIM / 64), loss_t);
    }
    finalize_kernel<<<1, 1, 0, stream>>>(loss_t, denom, out);
}